// HashEncoding_58926951301171
// MI455X (gfx1250) — compile-verified
//
#include <hip/hip_runtime.h>
#include <math.h>
#include <stdint.h>
#include <stddef.h>

// Instant-NGP multiresolution hash encoding, tuned for MI455X (gfx1250):
//  - 64 MB of hash tables are L2-resident (192 MB L2); kernel is gather-bound
//    (256M b64 gathers). VALU per gather is minimized so the CU's two SIMD32s
//    don't starve the shared memory path.
//  - Non-temporal hints on the 24 MB position stream and the 256 MB output
//    stream so neither evicts the tables from L2.
//  - Persistent blocks + global_prefetch_b8 of the next tile's positions.
//  - Positions staged through LDS with coalesced b128 NT loads; output
//    transposed through LDS so stores are fully-coalesced b128 NT stores.
//  - The reference's clip(floor+offset, 0, res) is a provable no-op for
//    positions in [-1,1) quantized to 2^-23 (floor in [0,res-1], corner+1<=res),
//    so corners use integer +1 and hy1 = hy0 + PRIME (u32 wraparound algebra),
//    eliminating ~220 VALU ops from the hot loop with bit-identical results.
//  - WMMA is not applicable: the 8x2 feats operand is private per point,
//    there is no shared matrix operand to feed v_wmma with.

typedef float v2f __attribute__((ext_vector_type(2)));
typedef float v4f __attribute__((ext_vector_type(4)));

#define NUM_LEVELS 16
#define TILE 256
#define HASH_SIZE (1u << 19)
#define HASH_MASK (HASH_SIZE - 1u)
#define PRIME_Y 2654435761u
#define PRIME_Z 805459861u

struct ResList { int r[NUM_LEVELS]; };

__global__ __launch_bounds__(TILE, 1) void hash_encode_kernel(
    const float* __restrict__ pos,      // [N,3]
    const float* __restrict__ tables,   // [16, 2^19, 2]
    float* __restrict__ out,            // [N, 32]
    int N, int numTiles, ResList res)
{
    // Output transpose buffer: 9-v4f stride per thread (8 used + 1 pad)
    // to spread LDS bank accesses.  36 KB of the WGP's 320 KB.
    __shared__ v4f   lds4[TILE * 9];
    // Position staging buffer: TILE*3 floats = 3 KB, loaded as 192 v4f.
    __shared__ float posBuf[TILE * 3];

    const int t = threadIdx.x;

    for (int tile = blockIdx.x; tile < numTiles; tile += gridDim.x) {
        const int base = tile * TILE;
        const int n = base + t;
        const bool fullTile = (base + TILE <= N);   // block-uniform

        // Software pipeline: prefetch the positions of this block's NEXT tile
        // while the gather storm for the current one is in flight.
        // Lowers to global_prefetch_b8 on gfx1250 (speculative; OOB dropped).
        const int nextTile = tile + gridDim.x;
        if (nextTile < numTiles) {
            __builtin_prefetch(pos + (size_t)(nextTile * TILE + t) * 3, 0, 3);
        }

        float nx = 0.f, ny = 0.f, nz = 0.f;
        if (fullTile) {
            // Coalesced, non-temporal staging of this tile's 3 KB of positions:
            // 192 b128 loads per block instead of 768 strided dword loads.
            __syncthreads();                   // posBuf reuse across iterations
            if (t < (TILE * 3) / 4) {
                const v4f* __restrict__ p4 =
                    (const v4f*)(pos + (size_t)base * 3);
                ((v4f*)posBuf)[t] = __builtin_nontemporal_load(p4 + t);
            }
            __syncthreads();
            // stride-3 dword reads: gcd(3,64)=1 -> bank-conflict free
            const float px = posBuf[t * 3 + 0];
            const float py = posBuf[t * 3 + 1];
            const float pz = posBuf[t * 3 + 2];
            nx = (px + 1.0f) * 0.5f;   // (p - BBOX_MIN)/(BBOX_MAX - BBOX_MIN)
            ny = (py + 1.0f) * 0.5f;
            nz = (pz + 1.0f) * 0.5f;
        } else if (n < N) {
            const float* p = pos + (size_t)n * 3;
            nx = (__builtin_nontemporal_load(p + 0) + 1.0f) * 0.5f;
            ny = (__builtin_nontemporal_load(p + 1) + 1.0f) * 0.5f;
            nz = (__builtin_nontemporal_load(p + 2) + 1.0f) * 0.5f;
        }

        const bool active = (n < N);
        v2f acc[NUM_LEVELS];

        #pragma unroll
        for (int l = 0; l < NUM_LEVELS; ++l) {
            v2f a = (v2f)(0.f);
            if (active) {
                const float fres = (float)res.r[l];
                const float sx = nx * fres, sy = ny * fres, sz = nz * fres;
                const float fx = floorf(sx), fy = floorf(sy), fz = floorf(sz);
                const float tx = sx - fx, ty = sy - fy, tz = sz - fz;

                // sx in [0, fres) exactly (see header comment), so
                // floor in [0, fres-1]: clip is a no-op, +1 is an int add.
                const uint32_t x0 = (uint32_t)fx;
                const uint32_t y0 = (uint32_t)fy;
                const uint32_t z0 = (uint32_t)fz;
                const uint32_t x1 = x0 + 1u;

                // Spatial hash (u32 wraparound), mod 2^19.  AND distributes
                // over XOR and x <= 512 < 2^19; (y+1)*P = y*P + P in u32.
                const uint32_t hy0u = y0 * PRIME_Y;
                const uint32_t hz0u = z0 * PRIME_Z;
                const uint32_t hy0 = hy0u & HASH_MASK;
                const uint32_t hy1 = (hy0u + PRIME_Y) & HASH_MASK;
                const uint32_t hz0 = hz0u & HASH_MASK;
                const uint32_t hz1 = (hz0u + PRIME_Z) & HASH_MASK;
                const uint32_t b00 = x0 ^ hy0, b01 = x0 ^ hy1;
                const uint32_t b10 = x1 ^ hy0, b11 = x1 ^ hy1;

                const v2f* __restrict__ tab =
                    (const v2f*)tables + (size_t)l * HASH_SIZE;

                // 8 independent b64 gathers (L2-resident) issued back to back.
                const v2f f000 = tab[b00 ^ hz0];
                const v2f f001 = tab[b00 ^ hz1];
                const v2f f010 = tab[b01 ^ hz0];
                const v2f f011 = tab[b01 ^ hz1];
                const v2f f100 = tab[b10 ^ hz0];
                const v2f f101 = tab[b10 ^ hz1];
                const v2f f110 = tab[b11 ^ hz0];
                const v2f f111 = tab[b11 ^ hz1];

                // Trilinear weights with explicit pair-product sharing
                // (same product algebra as the reference's einsum).
                const float wx0 = 1.f - tx, wy0 = 1.f - ty, wz0 = 1.f - tz;
                const float wxy00 = wx0 * wy0, wxy01 = wx0 * ty;
                const float wxy10 = tx  * wy0, wxy11 = tx  * ty;
                const float w000 = wxy00 * wz0, w001 = wxy00 * tz;
                const float w010 = wxy01 * wz0, w011 = wxy01 * tz;
                const float w100 = wxy10 * wz0, w101 = wxy10 * tz;
                const float w110 = wxy11 * wz0, w111 = wxy11 * tz;

                a.x = fmaf(w000, f000.x, fmaf(w001, f001.x,
                      fmaf(w010, f010.x, fmaf(w011, f011.x,
                      fmaf(w100, f100.x, fmaf(w101, f101.x,
                      fmaf(w110, f110.x, w111 * f111.x)))))));
                a.y = fmaf(w000, f000.y, fmaf(w001, f001.y,
                      fmaf(w010, f010.y, fmaf(w011, f011.y,
                      fmaf(w100, f100.y, fmaf(w101, f101.y,
                      fmaf(w110, f110.y, w111 * f111.y)))))));
            }
            acc[l] = a;
        }

        if (fullTile) {
            // Transpose through LDS so the 32 KB output tile is written with
            // fully-coalesced non-temporal b128 stores (512 B per wave-store).
            __syncthreads();   // protect lds4 reuse across tile iterations
            #pragma unroll
            for (int k = 0; k < 8; ++k) {
                v4f v;
                v.x = acc[2 * k].x;     v.y = acc[2 * k].y;
                v.z = acc[2 * k + 1].x; v.w = acc[2 * k + 1].y;
                lds4[t * 9 + k] = v;
            }
            __syncthreads();
            v4f* __restrict__ out4 = (v4f*)out + (size_t)base * 8;
            #pragma unroll
            for (int k = 0; k < 8; ++k) {
                const int j = k * TILE + t;               // v4f index in tile
                const v4f v = lds4[(j >> 3) * 9 + (j & 7)];
                // NT: 256 MB output stream must not evict the tables from L2.
                __builtin_nontemporal_store(v, out4 + j);
            }
        } else if (active) {
            // Tail tile (never hit for N = 2^21, kept for generality).
            v2f* __restrict__ o2 = (v2f*)out + (size_t)n * NUM_LEVELS;
            #pragma unroll
            for (int l = 0; l < NUM_LEVELS; ++l) {
                __builtin_nontemporal_store(acc[l], o2 + l);
            }
        }
    }
}

extern "C" void kernel_launch(void* const* d_in, const int* in_sizes, int n_in,
                              void* d_out, int out_size, void* d_ws, size_t ws_size,
                              hipStream_t stream) {
    const float* pos    = (const float*)d_in[0];   // [N,3] f32
    const float* tables = (const float*)d_in[1];   // [16, 2^19, 2] f32
    float* out          = (float*)d_out;           // [N, 32] f32

    const int N = in_sizes[0] / 3;

    // Replicate the reference's double-precision resolution formula exactly:
    // floor(BASE * exp(i * log(FINEST/BASE) / (L-1)))
    ResList res;
    for (int i = 0; i < NUM_LEVELS; ++i) {
        res.r[i] = (int)floor(16.0 * exp((double)i * log(32.0) / 15.0));
    }

    const int numTiles = (N + TILE - 1) / TILE;
    int grid = numTiles < 4096 ? numTiles : 4096;  // persistent blocks, >=2 tiles each
    if (grid < 1) grid = 1;

    hipLaunchKernelGGL(hash_encode_kernel, dim3(grid), dim3(TILE), 0, stream,
                       pos, tables, out, N, numTiles, res);
}